// FixedCFFsLayer_81793357185097
// MI455X (gfx1250) — compile-verified
//
#include <hip/hip_runtime.h>
#include <math.h>

// ---------------- types ----------------
typedef __attribute__((ext_vector_type(2))) float v2f;
typedef __attribute__((ext_vector_type(8))) float v8f;

__device__ __forceinline__ float frcp(float x) { return __builtin_amdgcn_rcpf(x); }
__device__ __forceinline__ float fsq(float x) { return x * x; }

#define PI_F 3.14159265358979f

// Physics constants
#define C_M     0.938272f
#define C_ALPHA (1.0f / 137.036f)
#define C_GEVNB 389379.36f
// vector / sea Regge params
#define C_ALV  0.43f
#define C_ALPV 0.85f
#define C_MV2  0.64f
#define C_BV   0.4f
#define C_KV   (1.35f * 0.918f)          /* NV*RV, (8/9+1/9)==1 */
#define C_ALS  1.13f
#define C_ALPS 0.15f
#define C_MS2  0.482f
#define C_KS   ((2.0f / 9.0f) * 1.5f * 1.0f) /* 2/9*NS*RS */
#define C_TMV2 (3.993f * 3.993f)
#define C_KTV  (0.6f * 0.881f)           /* NTV*TRV */
#define C_CSUB 2.768f
#define C_MC2  (1.204f * 1.204f)
#define C_RPI  2.646f
#define C_MPI2 16.0f

// ---------------------------------------------------------------------------
// Kernel 1: compute 64-pt Gauss-Legendre nodes/weights on [0,1] (Newton on
// P_64 in f64, matching numpy leggauss -> f32 cast), then fold all per-point
// constants. Layout in ws: 9 arrays of 64 floats:
//   [0]=x  [1]=w  [2]=log(twox)  [3]=Av  [4]=As  [5]=Atv  [6]=ov [7]=os [8]=ot
// ---------------------------------------------------------------------------
__global__ void gl_init_kernel(float* __restrict__ ws) {
  int i = threadIdx.x;
  if (i >= 64) return;
  double z = cos(3.14159265358979323846 * ((double)i + 0.75) / 64.5);
  double pp = 1.0;
  for (int it = 0; it < 24; ++it) {
    double p0 = 1.0, p1 = z;
    for (int j = 2; j <= 64; ++j) {
      double p2 = ((2.0 * j - 1.0) * z * p1 - (j - 1.0) * p0) / (double)j;
      p0 = p1;
      p1 = p2;
    }
    pp = 64.0 * (z * p1 - p0) / (z * z - 1.0);
    if (it < 23) z -= p1 / pp;  // last pass: leave z, keep converged pp
  }
  float x = (float)(0.5 * (z + 1.0));           // XG = 0.5*(z+1)
  float w = (float)(1.0 / ((1.0 - z * z) * pp * pp)); // WG = 0.5*wleg

  float twox = 2.0f * x / (1.0f + x);
  float onex = (1.0f - x) / (1.0f + x);
  float pio  = PI_F / (1.0f + x);
  float o04  = __powf(onex, C_BV);  // BV == TBV == 0.4

  ws[i + 0 * 64] = x;
  ws[i + 1 * 64] = w;
  ws[i + 2 * 64] = __logf(twox);
  ws[i + 3 * 64] = C_KV * o04 * pio;
  ws[i + 4 * 64] = C_KS * onex * onex * pio;
  ws[i + 5 * 64] = C_KTV * o04 * pio;
  ws[i + 6 * 64] = onex / C_MV2;
  ws[i + 7 * 64] = onex / C_MS2;
  ws[i + 8 * 64] = onex / C_TMV2;
}

// ---------------------------------------------------------------------------
// Kernel 2: main. 8 waves/block, 16 rows/wave. Quadrature summands feed
// V_WMMA_F32_16X16X4_F32 chains (B = ones) as K-accumulating reductions.
// ---------------------------------------------------------------------------
__global__ __launch_bounds__(256) void km15_kernel(const float* __restrict__ in,
                                                   const float* __restrict__ cons,
                                                   float* __restrict__ out,
                                                   int nrows) {
  __shared__ float c[576];
  __shared__ float red[8][4][16];  // [wave][sum][row]

  int tid = threadIdx.x;
  for (int idx = tid; idx < 576; idx += 256) c[idx] = cons[idx];
  __syncthreads();

  int lane = tid & 31;
  int wv   = tid >> 5;
  int r    = lane & 15;
  int row  = blockIdx.x * 128 + wv * 16 + r;
  if (row >= nrows) row = nrows - 1;  // tail clamp: duplicate identical work, EXEC stays full

  const float* ip = in + row * 5;
  float QQ = ip[0], xB = ip[1], t = ip[2], k = ip[4];

  float xi = xB * frcp(2.0f - xB);
  float pa = -C_ALV - C_ALPV * t;
  float ps = -C_ALS - C_ALPS * t;

  // Im parts at x = xi (also the ImH / ImHt CFFs)
  float twoxi = 2.0f * xi * frcp(1.0f + xi);
  float onexi = (1.0f - xi) * frcp(1.0f + xi);
  float Lxi   = __logf(twoxi);
  float pio   = PI_F * frcp(1.0f + xi);
  float oi04  = __powf(onexi, C_BV);
  float ev_i  = __expf(pa * Lxi);
  float es_i  = __expf(ps * Lxi);
  float imxi_h = pio * (C_KV * ev_i * oi04 * frcp(1.0f - onexi * t / C_MV2) +
                        C_KS * es_i * onexi * onexi * fsq(frcp(1.0f - onexi * t / C_MS2)));
  float imxi_t = pio * C_KTV * ev_i * oi04 * frcp(1.0f - onexi * t / C_TMV2);

  // ---- quadrature via WMMA K-accumulation --------------------------------
  v8f accPH = {0.f, 0.f, 0.f, 0.f, 0.f, 0.f, 0.f, 0.f};
  v8f accRH = {0.f, 0.f, 0.f, 0.f, 0.f, 0.f, 0.f, 0.f};
  v8f accPT = {0.f, 0.f, 0.f, 0.f, 0.f, 0.f, 0.f, 0.f};
  v8f accRT = {0.f, 0.f, 0.f, 0.f, 0.f, 0.f, 0.f, 0.f};
  v2f bones;
  bones.x = 1.0f;
  bones.y = 1.0f;

  int kbase = (lane >> 4) << 1;  // lanes 0-15 -> K{0,1}; lanes 16-31 -> K{2,3}

  for (int ch = 0; ch < 16; ++ch) {
    int j0 = ch * 4 + kbase;
    v2f aPH, aRH, aPT, aRT;
#pragma unroll
    for (int u = 0; u < 2; ++u) {
      int j = j0 + u;
      float x   = c[j];
      float wq  = c[j + 64];
      float L2  = c[j + 128];
      float Av  = c[j + 192];
      float As  = c[j + 256];
      float Atv = c[j + 320];
      float ov  = c[j + 384];
      float os  = c[j + 448];
      float ot  = c[j + 512];

      float ev = __expf(pa * L2);
      float es = __expf(ps * L2);
      float rv = frcp(1.0f - ov * t);
      float rs = frcp(1.0f - os * t);
      float rt = frcp(1.0f - ot * t);
      float imh = Av * ev * rv + As * es * rs * rs;
      float imt = Atv * ev * rt;
      float wr1 = wq * frcp(xi - x);
      float wr2 = wq * frcp(xi + x);
      aPH[u] = (imh - imxi_h) * wr1;
      aRH[u] = imh * wr2;
      aPT[u] = (imt - imxi_t) * wr1;
      aRT[u] = imt * wr2;
    }
    accPH = __builtin_amdgcn_wmma_f32_16x16x4_f32(false, aPH, false, bones, (short)0, accPH, false, false);
    accRH = __builtin_amdgcn_wmma_f32_16x16x4_f32(false, aRH, false, bones, (short)0, accRH, false, false);
    accPT = __builtin_amdgcn_wmma_f32_16x16x4_f32(false, aPT, false, bones, (short)0, accPT, false, false);
    accRT = __builtin_amdgcn_wmma_f32_16x16x4_f32(false, aRT, false, bones, (short)0, accRT, false, false);
  }

  // D layout: lanes 0-15 hold row sums 0-7 in v0..7; lanes 16-31 hold rows 8-15.
  if (lane == 0) {
#pragma unroll
    for (int v = 0; v < 8; ++v) {
      red[wv][0][v] = accPH[v];
      red[wv][1][v] = accRH[v];
      red[wv][2][v] = accPT[v];
      red[wv][3][v] = accRT[v];
    }
  }
  if (lane == 16) {
#pragma unroll
    for (int v = 0; v < 8; ++v) {
      red[wv][0][8 + v] = accPH[v];
      red[wv][1][8 + v] = accRH[v];
      red[wv][2][8 + v] = accPT[v];
      red[wv][3][8 + v] = accRT[v];
    }
  }
  __syncthreads();

  if (lane < 16) {
    float SPH = red[wv][0][r];
    float SRH = red[wv][1][r];
    float SPT = red[wv][2][r];
    float SRT = red[wv][3][r];

    float xlog = __logf(xi * frcp(1.0f - xi));
    float Dsub = C_CSUB * fsq(frcp(1.0f - t / C_MC2));
    float ReH  = (SPH + imxi_h * xlog - SRH) * (1.0f / PI_F) - Dsub;
    float ReE  = Dsub;
    float ReHt = (SPT + imxi_t * xlog - SRT) * (1.0f / PI_F);
    float ImH  = imxi_h;
    float ImHt = imxi_t;
    float ReEt = C_RPI * frcp(xi) * 2.164f * frcp(0.0196f - t) * fsq(frcp(1.0f - t / C_MPI2));

    float y  = QQ * frcp(2.0f * C_M * k * xB);
    float e2 = 4.0f * C_M * C_M * xB * xB * frcp(QQ);
    float HH   = ReH * ReH + ImH * ImH;
    float EE   = ReE * ReE;
    float HtHt = ReHt * ReHt + ImHt * ImHt;
    float EtEt = ReEt * ReEt;
    float HE   = ReH * ReE;
    float HtEt = ReHt * ReEt;
    float t4m2 = t * frcp(4.0f * C_M * C_M);
    float tw   = 2.0f - xB;
    float cdvcs = (4.0f * (1.0f - xB) * (HH + HtHt) - xB * xB * (2.0f * HE + 2.0f * HtEt) -
                   (xB * xB + tw * tw * t4m2) * EE - xB * xB * t4m2 * EtEt) * frcp(tw * tw);
    float c0   = 2.0f * (2.0f - 2.0f * y + y * y) * cdvcs;
    float amp2 = c0 * frcp(y * y * QQ);
    float a3   = C_ALPHA * C_ALPHA * C_ALPHA;
    float gmma = a3 * xB * y * frcp(16.0f * PI_F * PI_F * QQ * QQ * __builtin_amdgcn_sqrtf(1.0f + e2));
    float dvcs = gmma * amp2 * C_GEVNB;

    float* op = out + (size_t)row * 3;
    op[0] = ReE;
    op[1] = ReHt;
    op[2] = dvcs;
  }
}

extern "C" void kernel_launch(void* const* d_in, const int* in_sizes, int n_in,
                              void* d_out, int out_size, void* d_ws, size_t ws_size,
                              hipStream_t stream) {
  const float* in = (const float*)d_in[0];
  float* out = (float*)d_out;
  float* cons = (float*)d_ws;  // 576 floats = 2304 B
  int rows = in_sizes[0] / 5;

  hipLaunchKernelGGL(gl_init_kernel, dim3(1), dim3(64), 0, stream, cons);
  int nblk = (rows + 127) / 128;
  hipLaunchKernelGGL(km15_kernel, dim3(nblk), dim3(256), 0, stream, in, cons, out, rows);
}